// GAT_58128087384143
// MI455X (gfx1250) — compile-verified
//
#include <hip/hip_runtime.h>

typedef float    v2f  __attribute__((ext_vector_type(2)));
typedef float    v8f  __attribute__((ext_vector_type(8)));

#define LRELU_SLOPE 0.2f

// ---------- order-preserving float <-> uint encoding for atomic segment-max ----------
__device__ __forceinline__ unsigned fenc(float f) {
  unsigned u = __float_as_uint(f);
  return (u & 0x80000000u) ? ~u : (u | 0x80000000u);
}
__device__ __forceinline__ float fdec(unsigned u) {
  u = (u & 0x80000000u) ? (u & 0x7FFFFFFFu) : ~u;
  return __uint_as_float(u);
}

// =====================================================================
// GEMM: Hout[N,64] = X[N,K] @ Wm[K,64]   (K = 128 or 64), f32 WMMA.
// 256 threads = 8 waves; each wave computes a 16-row x 64-col tile.
// W staged TRANSPOSED in LDS (wldsT[col*stride + k], stride = K+2) so a
// B fragment is one contiguous 8B-aligned ds_load_b64. All 4 B fragments
// are fetched into distinct locals before the 4 WMMAs so the DS loads
// overlap instead of serializing on one register pair.
// =====================================================================
__global__ __launch_bounds__(256) void gat_gemm_wmma(
    const float* __restrict__ X, const float* __restrict__ Wm,
    float* __restrict__ Hout, int N, int K)
{
  __shared__ float wldsT[64 * 130];                // worst case K=128 -> stride 130
  const int stride = K + 2;                        // even -> 8B-aligned b64 loads
  const int tid = threadIdx.x;
  for (int i = tid; i < K * 64; i += 256) {
    const int k = i >> 6, col = i & 63;            // Wm is [K,64] row-major
    wldsT[col * stride + k] = Wm[i];
  }
  __syncthreads();

  const int wave = tid >> 5;
  const int lane = tid & 31;
  const int l16  = lane & 15;
  const int hsel = lane >> 4;                      // 0: lanes 0-15, 1: lanes 16-31
  const int rowBase = blockIdx.x * 128 + wave * 16;
  int row = rowBase + l16;
  if (row >= N) row = N - 1;                       // clamp loads; stores guarded

  v8f acc[4] = {};                                 // 4 N-tiles of 16 cols

  // A 16x4 layout: lanes 0-15 hold K=0,1 ; lanes 16-31 hold K=2,3.
  // B 4x16 layout mirrors it: lanes 0-15 hold rows K=0,1 ; lanes 16-31 K=2,3.
  const int koff = hsel * 2;
  const float* xrow = X + (size_t)row * K + koff;
  const float* wb0 = &wldsT[(0 * 16 + l16) * stride + koff];
  const float* wb1 = &wldsT[(1 * 16 + l16) * stride + koff];
  const float* wb2 = &wldsT[(2 * 16 + l16) * stride + koff];
  const float* wb3 = &wldsT[(3 * 16 + l16) * stride + koff];

  for (int kk = 0; kk < K; kk += 4) {
    const float2 av  = *(const float2*)(xrow + kk);  // global_load_b64
    const float2 bv0 = *(const float2*)(wb0 + kk);   // 4 independent ds_load_b64,
    const float2 bv1 = *(const float2*)(wb1 + kk);   // all in flight together
    const float2 bv2 = *(const float2*)(wb2 + kk);
    const float2 bv3 = *(const float2*)(wb3 + kk);
    v2f a;  a.x = av.x;   a.y = av.y;
    v2f b0; b0.x = bv0.x; b0.y = bv0.y;
    v2f b1; b1.x = bv1.x; b1.y = bv1.y;
    v2f b2; b2.x = bv2.x; b2.y = bv2.y;
    v2f b3; b3.x = bv3.x; b3.y = bv3.y;
    acc[0] = __builtin_amdgcn_wmma_f32_16x16x4_f32(false, a, false, b0, (short)0, acc[0], false, false);
    acc[1] = __builtin_amdgcn_wmma_f32_16x16x4_f32(false, a, false, b1, (short)0, acc[1], false, false);
    acc[2] = __builtin_amdgcn_wmma_f32_16x16x4_f32(false, a, false, b2, (short)0, acc[2], false, false);
    acc[3] = __builtin_amdgcn_wmma_f32_16x16x4_f32(false, a, false, b3, (short)0, acc[3], false, false);
  }

  // C/D layout: vgpr j -> rows (j, j+8) in lane-halves, col = lane&15 (+16*nt)
#pragma unroll
  for (int nt = 0; nt < 4; ++nt) {
#pragma unroll
    for (int j = 0; j < 8; ++j) {
      const int ro = rowBase + j + hsel * 8;
      if (ro < N) Hout[(size_t)ro * 64 + nt * 16 + l16] = acc[nt][j];
    }
  }
}

// =====================================================================
// Per-node attention logits: als[n] = h[n,:].a_src ; ald[n] = h[n,:].a_dst
// One wave32 per node; lane covers features f and f+32.
// =====================================================================
__global__ __launch_bounds__(256) void gat_logits(
    const float* __restrict__ H, const float* __restrict__ a_src,
    const float* __restrict__ a_dst, float* __restrict__ als,
    float* __restrict__ ald, int N)
{
  const int lane = threadIdx.x & 31;
  const int node = (int)((blockIdx.x * (unsigned)blockDim.x + threadIdx.x) >> 5);
  if (node >= N) return;
  const float h0 = H[(size_t)node * 64 + lane];
  const float h1 = H[(size_t)node * 64 + lane + 32];
  float ps = h0 * a_src[lane] + h1 * a_src[lane + 32];
  float pd = h0 * a_dst[lane] + h1 * a_dst[lane + 32];
#pragma unroll
  for (int o = 16; o > 0; o >>= 1) {
    ps += __shfl_xor(ps, o, 32);
    pd += __shfl_xor(pd, o, 32);
  }
  if (lane == 0) { als[node] = ps; ald[node] = pd; }
}

// Zero agg, denom; init segment-max to encoded -inf (0u).
__global__ void gat_init(unsigned* __restrict__ mEnc, float* __restrict__ denom,
                         float* __restrict__ agg, int N)
{
  const int i = blockIdx.x * blockDim.x + threadIdx.x;
  if (i < N) { mEnc[i] = 0u; denom[i] = 0.0f; }
  if (i < N * 64) agg[i] = 0.0f;
}

// Pass 1: e = leaky_relu(als[src]+ald[dst]); atomic segment max over dst.
__global__ void gat_edge_logit_max(
    const int* __restrict__ ei, int E, int N,
    const float* __restrict__ als, const float* __restrict__ ald,
    float* __restrict__ ebuf, unsigned* __restrict__ mEnc)
{
  const int e = blockIdx.x * blockDim.x + threadIdx.x;
  if (e >= E + N) return;
  int s, d;
  if (e < E) { s = ei[e]; d = ei[E + e]; } else { s = d = e - E; }
  float v = als[s] + ald[d];
  v = (v > 0.0f) ? v : LRELU_SLOPE * v;
  ebuf[e] = v;
  atomicMax(&mEnc[d], fenc(v));
}

// Pass 2: ex = exp(e - m[dst]); atomic segment sum over dst.
__global__ void gat_edge_exp_sum(
    const int* __restrict__ ei, int E, int N,
    const float* __restrict__ ebuf, const unsigned* __restrict__ mEnc,
    float* __restrict__ exbuf, float* __restrict__ denom)
{
  const int e = blockIdx.x * blockDim.x + threadIdx.x;
  if (e >= E + N) return;
  int d;
  if (e < E) { d = ei[E + e]; } else { d = e - E; }
  const float ex = __expf(ebuf[e] - fdec(mEnc[d]));
  exbuf[e] = ex;
  atomicAdd(&denom[d], ex);
}

// Pass 3: wave32 per edge: agg[dst,:] += alpha * h[src,:]
// Lane owns features {2*lane, 2*lane+1}: one global_load_b64 gather/lane.
__global__ __launch_bounds__(256) void gat_edge_scatter(
    const int* __restrict__ ei, int E, int N,
    const float* __restrict__ Hin, const float* __restrict__ exbuf,
    const float* __restrict__ denom, float* __restrict__ agg)
{
  const int lane = threadIdx.x & 31;
  const long long g = (long long)blockIdx.x * blockDim.x + threadIdx.x;
  const int e = (int)(g >> 5);
  if (e >= E + N) return;
  int s, d;
  if (e < E) { s = ei[e]; d = ei[E + e]; } else { s = d = e - E; }
  const float alpha = exbuf[e] / denom[d];
  const float2 hv = *(const float2*)&Hin[(size_t)s * 64 + lane * 2]; // b64 gather
  atomicAdd(&agg[(size_t)d * 64 + lane * 2 + 0], alpha * hv.x);
  atomicAdd(&agg[(size_t)d * 64 + lane * 2 + 1], alpha * hv.y);
}

// Epilogue: out = agg + bias (optional ReLU)
__global__ void gat_epilogue(const float* __restrict__ agg,
                             const float* __restrict__ bias,
                             float* __restrict__ out, int N, int doRelu)
{
  const int i = blockIdx.x * blockDim.x + threadIdx.x;
  if (i >= N * 64) return;
  float v = agg[i] + bias[i & 63];
  if (doRelu) v = fmaxf(v, 0.0f);
  out[i] = v;
}

// =====================================================================
extern "C" void kernel_launch(void* const* d_in, const int* in_sizes, int n_in,
                              void* d_out, int out_size, void* d_ws, size_t ws_size,
                              hipStream_t stream) {
  const float* x   = (const float*)d_in[0];
  const int*   ei  = (const int*)  d_in[1];
  const float* W1  = (const float*)d_in[2];
  const float* as1 = (const float*)d_in[3];
  const float* ad1 = (const float*)d_in[4];
  const float* b1  = (const float*)d_in[5];
  const float* W2  = (const float*)d_in[6];
  const float* as2 = (const float*)d_in[7];
  const float* ad2 = (const float*)d_in[8];
  const float* b2  = (const float*)d_in[9];

  const int N  = in_sizes[0] / 128;   // 50000
  const int E  = in_sizes[1] / 2;     // 800000
  const int ET = E + N;               // with self-loops

  // workspace layout
  char* w = (char*)d_ws;
  float*    h1    = (float*)w;    w += (size_t)N * 64 * sizeof(float);
  float*    h2    = (float*)w;    w += (size_t)N * 64 * sizeof(float);
  float*    agg   = (float*)w;    w += (size_t)N * 64 * sizeof(float);
  float*    als   = (float*)w;    w += (size_t)N * sizeof(float);
  float*    ald   = (float*)w;    w += (size_t)N * sizeof(float);
  unsigned* mEnc  = (unsigned*)w; w += (size_t)N * sizeof(unsigned);
  float*    denom = (float*)w;    w += (size_t)N * sizeof(float);
  float*    ebuf  = (float*)w;    w += (size_t)ET * sizeof(float);
  float*    exbuf = (float*)w;    w += (size_t)ET * sizeof(float);

  const dim3 blk(256);
  const int gGemm     = (N + 127) / 128;
  const int gNode     = (N * 64 + 255) / 256;
  const int gWarpNode = (N * 32 + 255) / 256;
  const int gEdge     = (ET + 255) / 256;
  const int gWaveEdge = (int)(((long long)ET * 32 + 255) / 256);

  // ---------------- Layer 1 ----------------
  gat_gemm_wmma    <<<gGemm,     blk, 0, stream>>>(x,  W1, h1, N, 128);
  gat_logits       <<<gWarpNode, blk, 0, stream>>>(h1, as1, ad1, als, ald, N);
  gat_init         <<<gNode,     blk, 0, stream>>>(mEnc, denom, agg, N);
  gat_edge_logit_max<<<gEdge,    blk, 0, stream>>>(ei, E, N, als, ald, ebuf, mEnc);
  gat_edge_exp_sum <<<gEdge,     blk, 0, stream>>>(ei, E, N, ebuf, mEnc, exbuf, denom);
  gat_edge_scatter <<<gWaveEdge, blk, 0, stream>>>(ei, E, N, h1, exbuf, denom, agg);
  gat_epilogue     <<<gNode,     blk, 0, stream>>>(agg, b1, h2, N, 1);   // h2 = relu(conv1)

  // ---------------- Layer 2 ----------------
  gat_gemm_wmma    <<<gGemm,     blk, 0, stream>>>(h2, W2, h1, N, 64);   // h1 reused: layer-2 features
  gat_logits       <<<gWarpNode, blk, 0, stream>>>(h1, as2, ad2, als, ald, N);
  gat_init         <<<gNode,     blk, 0, stream>>>(mEnc, denom, agg, N);
  gat_edge_logit_max<<<gEdge,    blk, 0, stream>>>(ei, E, N, als, ald, ebuf, mEnc);
  gat_edge_exp_sum <<<gEdge,     blk, 0, stream>>>(ei, E, N, ebuf, mEnc, exbuf, denom);
  gat_edge_scatter <<<gWaveEdge, blk, 0, stream>>>(ei, E, N, h1, exbuf, denom, agg);
  gat_epilogue     <<<gNode,     blk, 0, stream>>>(agg, b2, (float*)d_out, N, 0);
}